// DisentangleGraph_7035156431055
// MI455X (gfx1250) — compile-verified
//
#include <hip/hip_runtime.h>
#include <hip/hip_bf16.h>

// Problem constants (match reference)
constexpr int B_   = 128;
constexpr int N_   = 512;
constexpr int NE_  = 512;
constexpr int K_   = 8;
constexpr int DIM_ = 256;
constexpr int OUTC = K_ + NE_;          // 520
#define E_RATIO 0.3f
#define TEMPV   10.0f
#define EPSV    1e-8f

typedef __attribute__((ext_vector_type(2))) float v2f;
typedef __attribute__((ext_vector_type(8))) float v8f;

// ---------------------------------------------------------------------------
// Row L2 norms: one wave32 per row of a (rows, DIM_) matrix.
// ---------------------------------------------------------------------------
__global__ void row_norm_kernel(const float* __restrict__ x,
                                float* __restrict__ nrm, int rows) {
    int wave = threadIdx.x >> 5;
    int lane = threadIdx.x & 31;
    int row  = blockIdx.x * 8 + wave;
    if (row >= rows) return;
    const float* p = x + (size_t)row * DIM_;
    float s = 0.f;
#pragma unroll
    for (int i = 0; i < DIM_ / 32; ++i) {
        float v = p[lane + i * 32];
        s += v * v;
    }
#pragma unroll
    for (int m = 16; m >= 1; m >>= 1) s += __shfl_xor(s, m, 32);
    if (lane == 0) nrm[row] = sqrtf(s);
}

// ---------------------------------------------------------------------------
// select_k[b] = floor(E_RATIO * sum(mask[b,:])) ; one block per batch.
// ---------------------------------------------------------------------------
__global__ void selectk_kernel(const int* __restrict__ mask,
                               float* __restrict__ selk) {
    __shared__ float red[N_];
    int b = blockIdx.x, i = threadIdx.x;
    red[i] = (float)mask[b * N_ + i];
    __syncthreads();
    for (int s = N_ / 2; s >= 1; s >>= 1) {
        if (i < s) red[i] += red[i + s];
        __syncthreads();
    }
    if (i == 0) selk[b] = floorf(E_RATIO * red[0]);
}

// ---------------------------------------------------------------------------
// dots[b,n,k] = sum_d hidden[b,n,d] * int_emb[k,d]  via V_WMMA_F32_16X16X4_F32
// One wave32 per (b, 16-node tile). Cols 8..15 of the B operand are zero pad.
// A frag (16x4 f32): lane l<16 -> M=l, v0=K0,v1=K1 ; lane l>=16 -> K2,K3.
// B frag (4x16 f32): mirrored (lane&15 = N col, half selects K pair).
// C/D (16x16 f32):   lane&15 = N col, VGPR r -> M=r (+8 for lanes>=16).
// ---------------------------------------------------------------------------
__global__ void wmma_dots_kernel(const float* __restrict__ hidden,
                                 const float* __restrict__ emb,
                                 float* __restrict__ dots) {
    __shared__ float ldsA[16 * DIM_];   // 16 hidden rows (contiguous in memory)
    __shared__ float ldsB[16 * DIM_];   // int_emb rows 0..7, rows 8..15 zero
    const int lane = threadIdx.x;       // blockDim.x == 32
    const int b    = blockIdx.y;
    const int n0   = blockIdx.x * 16;

    // Cooperative, fully-coalesced staging into LDS (float4 granularity).
    const float4* h4 = (const float4*)(hidden + ((size_t)b * N_ + n0) * DIM_);
    const float4* e4 = (const float4*)emb;
    float4* a4 = (float4*)ldsA;
    float4* b4 = (float4*)ldsB;
    const float4 z4 = {0.f, 0.f, 0.f, 0.f};
#pragma unroll 4
    for (int idx = lane; idx < 16 * DIM_ / 4; idx += 32) {
        a4[idx] = h4[idx];
        b4[idx] = (idx < K_ * DIM_ / 4) ? e4[idx] : z4;
    }
    __syncthreads();

    const int row  = lane & 15;   // A: M row ; B: N col
    const int half = lane >> 4;   // selects K pair {0,1} vs {2,3}
    v8f c = {};
#pragma unroll 8
    for (int kk = 0; kk < DIM_; kk += 4) {
        const int o = kk + 2 * half;
        v2f a  = *(const v2f*)(ldsA + row * DIM_ + o);
        v2f bm = *(const v2f*)(ldsB + row * DIM_ + o);
        c = __builtin_amdgcn_wmma_f32_16x16x4_f32(
                /*neg_a=*/false, a, /*neg_b=*/false, bm,
                /*c_mod=*/(short)0, c, /*reuse_a=*/false, /*reuse_b=*/false);
    }

    // Write back valid columns (k < 8). lane&15 = k, VGPR r -> node row.
    if (row < K_) {
#pragma unroll
        for (int r = 0; r < 8; ++r) {
            int m = r + 8 * half;
            dots[((size_t)b * N_ + n0 + m) * K_ + row] = c[r];
        }
    }
}

// ---------------------------------------------------------------------------
// Stable-descending rank along the node dim, thresholded at select_k.
// rank(i) = #{j: s_j > s_i} + #{j<i: s_j == s_i}   (== double-argsort rank)
// One block per (b,k); 512 threads; sims broadcast from LDS.
// ---------------------------------------------------------------------------
__global__ void rank_kernel(const float* __restrict__ dots,
                            const float* __restrict__ nodenorm,
                            const float* __restrict__ intnorm,
                            const float* __restrict__ selk,
                            const int* __restrict__ mask,
                            float* __restrict__ out) {
    __shared__ float s[N_];
    const int k = blockIdx.x, b = blockIdx.y, i = threadIdx.x;

    float d     = dots[((size_t)b * N_ + i) * K_ + k];
    float denom = fmaxf(nodenorm[b * N_ + i] * intnorm[k], EPSV);
    float sim   = (TEMPV * d / denom) * (float)mask[b * N_ + i];
    s[i] = sim;
    __syncthreads();

    const float si = s[i];
    int cnt = 0;
    for (int j = 0; j < N_; ++j) {
        float sj = s[j];
        cnt += (sj > si) ? 1 : ((sj == si && j < i) ? 1 : 0);
    }
    out[((size_t)b * N_ + i) * OUTC + k] =
        ((float)cnt <= selk[b]) ? 3.0f : 0.0f;
}

// ---------------------------------------------------------------------------
// Bandwidth pass: copy H (B,N,512) into out columns 8..519, float4-vectorized.
// out row stride 520 floats; offset 8 floats -> both 16-byte aligned.
// ---------------------------------------------------------------------------
__global__ void copy_edges_kernel(const float4* __restrict__ H4,
                                  float4* __restrict__ out4) {
    size_t idx = (size_t)blockIdx.x * blockDim.x + threadIdx.x; // B*N*128 total
    size_t row = idx >> 7;       // /128 f4 per row
    size_t e   = idx & 127;
    out4[row * (OUTC / 4) + (K_ / 4) + e] = H4[idx];
}

// ---------------------------------------------------------------------------
extern "C" void kernel_launch(void* const* d_in, const int* in_sizes, int n_in,
                              void* d_out, int out_size, void* d_ws, size_t ws_size,
                              hipStream_t stream) {
    const float* hidden  = (const float*)d_in[0];   // (B,N,DIM)
    const float* H       = (const float*)d_in[1];   // (B,N,NE)
    const float* int_emb = (const float*)d_in[2];   // (K,DIM)
    const int*   mask    = (const int*)d_in[3];     // (B,N)
    float* out = (float*)d_out;                     // (B,N,520)

    // Workspace carve-up (~2.4 MB total)
    float* dots     = (float*)d_ws;                    // B*N*K
    float* nodenorm = dots + (size_t)B_ * N_ * K_;     // B*N
    float* intnorm  = nodenorm + (size_t)B_ * N_;      // K
    float* selk     = intnorm + K_;                    // B

    // Norms: 65536 node rows (8 per 256-thread block) + 8 emb rows (1 block)
    row_norm_kernel<<<(B_ * N_) / 8, 256, 0, stream>>>(hidden, nodenorm, B_ * N_);
    row_norm_kernel<<<1, 256, 0, stream>>>(int_emb, intnorm, K_);

    // Per-batch select_k
    selectk_kernel<<<B_, N_, 0, stream>>>(mask, selk);

    // WMMA einsum: (N/16) x B tiles, one wave32 each
    wmma_dots_kernel<<<dim3(N_ / 16, B_), 32, 0, stream>>>(hidden, int_emb, dots);

    // Rank + threshold -> out[:, :, 0:8]
    rank_kernel<<<dim3(K_, B_), N_, 0, stream>>>(dots, nodenorm, intnorm, selk,
                                                 mask, out);

    // H passthrough -> out[:, :, 8:520]
    const size_t total4 = (size_t)B_ * N_ * (NE_ / 4);   // 8388608
    copy_edges_kernel<<<(unsigned)(total4 / 256), 256, 0, stream>>>(
        (const float4*)H, (float4*)out);
}